// GraphNorm_6734508720712
// MI455X (gfx1250) — compile-verified
//
#include <hip/hip_runtime.h>
#include <hip/hip_bf16.h>
#include <stdint.h>

#define N_NODES   100000
#define N_GRAPHS  512
#define N_CHANNEL 128
#define COLS_TOTAL 1664   // 128 + 384 + 1152
#define EPSF 0.001f
#define STAGE_ROWS 16     // rows per async stage per wave

typedef float v4f __attribute__((ext_vector_type(4)));

// ---------------------------------------------------------------------------
// Stage 1: find per-graph node ranges (batch is sorted, every graph present)
// ---------------------------------------------------------------------------
__global__ void k_starts(const long long* __restrict__ batch,
                         int* __restrict__ starts, int n) {
    int i = blockIdx.x * 256 + threadIdx.x;
    if (i == 0) starts[N_GRAPHS] = n;
    if (i < n) {
        int g = (int)batch[i];
        if (i == 0 || (int)batch[i - 1] != g) starts[g] = i;
    }
}

// ---------------------------------------------------------------------------
// Stage 2: per-(graph, column) sum & sum-of-squares.
// Block = 128 threads = 4 waves; each wave owns a 32-column slice and
// double-buffers STAGE_ROWS rows/stage into LDS via
// global_load_async_to_lds_b128 (ASYNCcnt-tracked, NT-hinted).
// grid = (S/128, N_GRAPHS); S is always a multiple of 128 -> no masking.
// ---------------------------------------------------------------------------
__global__ void k_reduce(const float* __restrict__ X,
                         const int* __restrict__ starts,
                         float* __restrict__ sumbuf,
                         float* __restrict__ sqbuf,
                         int S, int wayOff) {
    __shared__ float lds[4][2][STAGE_ROWS][32];  // 16 KB / block

    const int g    = blockIdx.y;
    const int tid  = threadIdx.x;
    const int w    = tid >> 5;
    const int lane = tid & 31;
    const int jblk = blockIdx.x * 128;
    const int j    = jblk + tid;                 // this thread's column

    const int ns = starts[g];
    const int ne = starts[g + 1];

    // lane -> (row-within-quad, 4-float column offset) for 16B async copies
    const int rsub = lane >> 3;                  // 0..3
    const int coff = (lane & 7) * 4;             // 0,4,...,28 (floats)
    const long long colbase = (long long)jblk + w * 32 + coff;

    const unsigned ldsBase = (unsigned)(size_t)&lds[w][0][0][0];

    auto issue = [&](int bufi, int n0) {
#pragma unroll
        for (int i = 0; i < STAGE_ROWS / 4; ++i) {   // 4 x (4 rows * 128B)
            int row = n0 + i * 4 + rsub;
            if (row > ne - 1) row = ne - 1;          // clamp; surplus unread
            const float* gp = X + (long long)row * S + colbase;
            unsigned lo = ldsBase + (unsigned)bufi * (STAGE_ROWS * 128u)
                        + (unsigned)(i * 4 + rsub) * 128u + (unsigned)coff * 4u;
            asm volatile("global_load_async_to_lds_b128 %0, %1, off th:TH_LOAD_NT"
                         :: "v"(lo), "v"((unsigned long long)(size_t)gp)
                         : "memory");
        }
    };

    float sum = 0.f, sq = 0.f;
    int buf = 0;
    issue(0, ns);
    for (int n0 = ns; n0 < ne; n0 += STAGE_ROWS) {
        int n1 = n0 + STAGE_ROWS;
        if (n1 < ne) {                           // overlap next stage's copy
            issue(buf ^ 1, n1);
            asm volatile("s_wait_asynccnt 0x4" ::: "memory");
        } else {
            asm volatile("s_wait_asynccnt 0x0" ::: "memory");
        }
        int rows = ne - n0;
        if (rows >= STAGE_ROWS) {
#pragma unroll
            for (int r = 0; r < STAGE_ROWS; ++r) {
                float v = lds[w][buf][r][lane];
                sum += v;
                sq = fmaf(v, v, sq);
            }
        } else {
            for (int r = 0; r < rows; ++r) {
                float v = lds[w][buf][r][lane];
                sum += v;
                sq = fmaf(v, v, sq);
            }
        }
        buf ^= 1;
    }

    const long long o = (long long)g * COLS_TOTAL + wayOff + j;
    sumbuf[o] = sum;
    sqbuf[o]  = sq;
}

// ---------------------------------------------------------------------------
// Stage 3: turn (sum, sumsq) into per-(graph,column) scale/shift, in place.
//   mean = sum/deg ; m = mean*alpha ; var = sumsq/deg - 2*m*mean + m^2
//   s = gamma * rsqrt(var+eps) ; b = -m*s (+ beta for way 0)
// ---------------------------------------------------------------------------
__global__ void k_finalize(float* __restrict__ A,  // in: sum   out: s
                           float* __restrict__ B,  // in: sumsq out: b
                           const float* __restrict__ alpha,
                           const float* __restrict__ beta,
                           const float* __restrict__ gamma,
                           const float* __restrict__ degree) {
    unsigned i   = blockIdx.x * 256 + threadIdx.x;     // 0 .. 512*1664-1 exact
    unsigned g   = i / COLS_TOTAL;
    unsigned col = i % COLS_TOTAL;
    int way, c;
    if (col < 128)      { way = 0; c = (int)col; }
    else if (col < 512) { way = 1; c = (int)(col - 128) / 3; }
    else                { way = 2; c = (int)(col - 512) / 9; }

    float invdeg = 1.0f / degree[g];
    float mean = A[i] * invdeg;
    float m    = mean * alpha[way * N_CHANNEL + c];
    float var  = fmaf(B[i], invdeg, fmaf(m, m, -2.0f * m * mean));
    float s    = gamma[way * N_CHANNEL + c] * rsqrtf(var + EPSF);
    float b    = -m * s + (way == 0 ? beta[c] : 0.0f);
    A[i] = s;
    B[i] = b;
}

// ---------------------------------------------------------------------------
// Stage 4: streaming normalize: out = x*s + b. Non-temporal 128-bit traffic
// (working set >> 192MB L2); coefficient vectors stay cache-resident.
// Element counts divide 1024 exactly -> no bounds checks.
// ---------------------------------------------------------------------------
template <int S>
__global__ void k_apply(const float* __restrict__ X,
                        float* __restrict__ OUT,
                        const long long* __restrict__ batch,
                        const float* __restrict__ coefS,
                        const float* __restrict__ coefB,
                        int wayOff) {
    unsigned e = (blockIdx.x * 256u + threadIdx.x) * 4u;   // < 2^32
    unsigned n = e / (unsigned)S;                          // const-div -> mulhi
    unsigned j = e - n * (unsigned)S;
    int g = (int)batch[n];

    v4f x = __builtin_nontemporal_load((const v4f*)(X + e));
    unsigned ci = (unsigned)g * COLS_TOTAL + (unsigned)wayOff + j; // 16B aligned
    v4f s = *(const v4f*)(coefS + ci);
    v4f b = *(const v4f*)(coefB + ci);
    v4f o = x * s + b;
    __builtin_nontemporal_store(o, (v4f*)(OUT + e));
}

// ---------------------------------------------------------------------------
extern "C" void kernel_launch(void* const* d_in, const int* in_sizes, int n_in,
                              void* d_out, int out_size, void* d_ws, size_t ws_size,
                              hipStream_t stream) {
    const float*     x0     = (const float*)d_in[0];
    const float*     x1     = (const float*)d_in[1];
    const float*     x2     = (const float*)d_in[2];
    const float*     alpha  = (const float*)d_in[3];
    const float*     beta   = (const float*)d_in[4];
    const float*     gamma  = (const float*)d_in[5];
    const long long* batch  = (const long long*)d_in[6];
    const float*     degree = (const float*)d_in[7];
    float*           out    = (float*)d_out;

    // workspace layout (~6.82 MB)
    float* sumbuf = (float*)d_ws;                        // 512*1664
    float* sqbuf  = sumbuf + (size_t)N_GRAPHS * COLS_TOTAL;
    int*   starts = (int*)(sqbuf + (size_t)N_GRAPHS * COLS_TOTAL); // 513 ints

    k_starts<<<(N_NODES + 255) / 256, 256, 0, stream>>>(batch, starts, N_NODES);

    dim3 blk(128);
    k_reduce<<<dim3(1, N_GRAPHS), blk, 0, stream>>>(x0, starts, sumbuf, sqbuf, 128, 0);
    k_reduce<<<dim3(3, N_GRAPHS), blk, 0, stream>>>(x1, starts, sumbuf, sqbuf, 384, 128);
    k_reduce<<<dim3(9, N_GRAPHS), blk, 0, stream>>>(x2, starts, sumbuf, sqbuf, 1152, 512);

    k_finalize<<<(N_GRAPHS * COLS_TOTAL) / 256, 256, 0, stream>>>(
        sumbuf, sqbuf, alpha, beta, gamma, degree);

    const long long E0 = (long long)N_NODES * 128;       // 12.8M
    const long long E1 = (long long)N_NODES * 384;       // 38.4M
    k_apply<128> <<<(unsigned)(E0 / 1024), 256, 0, stream>>>(x0, out,           batch, sumbuf, sqbuf, 0);
    k_apply<384> <<<(unsigned)(E1 / 1024), 256, 0, stream>>>(x1, out + E0,      batch, sumbuf, sqbuf, 128);
    k_apply<1152><<<(unsigned)((long long)N_NODES * 1152 / 1024), 256, 0, stream>>>(
                                                          x2, out + E0 + E1,  batch, sumbuf, sqbuf, 512);
}